// SegmentLoss_72834055406384
// MI455X (gfx1250) — compile-verified
//
#include <hip/hip_runtime.h>
#include <math.h>

typedef __attribute__((ext_vector_type(2))) float v2f;
typedef __attribute__((ext_vector_type(8))) float v8f;

#define MAXB 256
#define TPB 256
#define BLOCKS 512
#define WAVES_PER_BLOCK (TPB / 32)
#define NPART (BLOCKS * WAVES_PER_BLOCK)

// ---------------------------------------------------------------------------
// Kernel 1: per-graph histogram + exclusive prefix (batch is sorted, so each
// graph's rows/cols form one contiguous range).
// ---------------------------------------------------------------------------
__global__ void k_counts(const int* __restrict__ batch, int n,
                         const int* __restrict__ nbp,
                         int* __restrict__ counts, int* __restrict__ starts) {
  __shared__ int sc[MAXB];
  for (int i = threadIdx.x; i < MAXB; i += blockDim.x) sc[i] = 0;
  __syncthreads();
  for (int i = threadIdx.x; i < n; i += blockDim.x) {
    atomicAdd(&sc[batch[i] & (MAXB - 1)], 1);
  }
  __syncthreads();
  if (threadIdx.x == 0) {
    const int nb = *nbp;
    int acc = 0;
    for (int b = 0; b < nb; ++b) {
      starts[b] = acc;
      counts[b] = sc[b];
      acc += sc[b];
    }
  }
}

// Branchless, numerically stable binary focal element (gamma == 2):
//   log p     = min(x,0)  - log1p(e^{-|x|})
//   log(1-p)  = min(-x,0) - log1p(e^{-|x|})
//   focal     = -( y*(1-p)^2*log p + (1-y)*p^2*log(1-p) )
__device__ __forceinline__ float focal_elem(float x, int yv) {
  const float e = __expf(-fabsf(x));
  const float l1pe = log1pf(e);
  const float logp = fminf(x, 0.0f) - l1pe;
  const float log1mp = fminf(-x, 0.0f) - l1pe;
  const float rcp = 1.0f / (1.0f + e);
  const float p = (x >= 0.0f) ? rcp : e * rcp;   // sigmoid(x), cndmask not branch
  const float omp = 1.0f - p;
  const float yf = (float)yv;                    // 0.0 or 1.0
  return -(yf * (omp * omp) * logp + (1.0f - yf) * (p * p) * log1mp);
}

// ---------------------------------------------------------------------------
// Kernel 2: stream ONLY the block-diagonal columns of each row (16x less HBM
// traffic than the dense N^2 read), branchless focal math, and reduce each
// wave's 32 lane accumulators with a single V_WMMA_F32_16X16X4_F32
// (A = lane accumulators, B = ones  =>  D[m,n] = acc[m] + acc[m+16]).
// ---------------------------------------------------------------------------
__global__ void k_focal(const float* __restrict__ pred,
                        const int* __restrict__ ylab,
                        const int* __restrict__ batch,
                        const int* __restrict__ counts,
                        const int* __restrict__ starts,
                        float* __restrict__ partials, int n) {
  const int lane = threadIdx.x & 31;
  const int gwave = blockIdx.x * WAVES_PER_BLOCK + (threadIdx.x >> 5);
  const int nwaves = gridDim.x * WAVES_PER_BLOCK;

  float acc = 0.0f;
  for (int r = gwave; r < n; r += nwaves) {
    // Row-uniform scalars -> SGPRs (r is wave-uniform by construction).
    const int b = __builtin_amdgcn_readfirstlane(batch[r]);
    const int s = __builtin_amdgcn_readfirstlane(starts[b]);
    const int cnt = __builtin_amdgcn_readfirstlane(counts[b]);
    const float inv = 1.0f / ((float)cnt * (float)cnt);
    const float* __restrict__ prow = pred + (size_t)r * (size_t)n + s;
    const int* __restrict__ yrow = ylab + (size_t)r * (size_t)n + s;

    float rsum = 0.0f;
    int i = lane;
    // 4x unrolled body: 8 outstanding loads + independent transcendentals.
    for (; i + 96 < cnt; i += 128) {
      const float x0 = prow[i];      const int y0 = yrow[i];
      const float x1 = prow[i + 32]; const int y1 = yrow[i + 32];
      const float x2 = prow[i + 64]; const int y2 = yrow[i + 64];
      const float x3 = prow[i + 96]; const int y3 = yrow[i + 96];
      rsum += focal_elem(x0, y0);
      rsum += focal_elem(x1, y1);
      rsum += focal_elem(x2, y2);
      rsum += focal_elem(x3, y3);
    }
    for (; i < cnt; i += 32) {
      rsum += focal_elem(prow[i], yrow[i]);
    }
    acc += rsum * inv;
  }

  // ---- wave reduction via WMMA (EXEC all-ones here: loops reconverged,
  // block fully populated). A(16x4): lanes0-15 -> K=0, lanes16-31 -> K=2.
  v2f a;  a[0] = acc;  a[1] = 0.0f;
  v2f bo; bo[0] = 1.0f; bo[1] = 1.0f;          // all-ones B is layout-invariant
  v8f c = {};
  c = __builtin_amdgcn_wmma_f32_16x16x4_f32(
      /*neg_a=*/false, a, /*neg_b=*/false, bo,
      /*c_mod=*/(short)0, c, /*reuse_a=*/false, /*reuse_b=*/false);
  // lane L<16 holds D[0..7][L] = s[0..7]; lane L>=16 holds s[8..15]
  float t = c[0] + c[1] + c[2] + c[3] + c[4] + c[5] + c[6] + c[7];
  t += __shfl_xor(t, 16, 32);                  // total wave sum in all lanes
  if (lane == 0) partials[gwave] = t;
}

// ---------------------------------------------------------------------------
// Kernel 3: deterministic fixed-order reduction of wave partials, divide by B.
// ---------------------------------------------------------------------------
__global__ void k_final(const float* __restrict__ partials, int np,
                        const int* __restrict__ nbp, float* __restrict__ out) {
  __shared__ float sm[TPB];
  float s = 0.0f;
  for (int i = threadIdx.x; i < np; i += blockDim.x) s += partials[i];
  sm[threadIdx.x] = s;
  __syncthreads();
  for (int off = TPB / 2; off > 0; off >>= 1) {
    if ((int)threadIdx.x < off) sm[threadIdx.x] += sm[threadIdx.x + off];
    __syncthreads();
  }
  if (threadIdx.x == 0) out[0] = sm[0] / (float)(*nbp);
}

extern "C" void kernel_launch(void* const* d_in, const int* in_sizes, int n_in,
                              void* d_out, int out_size, void* d_ws, size_t ws_size,
                              hipStream_t stream) {
  const float* pred = (const float*)d_in[0];   // [N, N] float32
  const int* ylab = (const int*)d_in[1];       // [N, N] int32
  const int* batch = (const int*)d_in[2];      // [N] int32 (sorted)
  const int* nbp = (const int*)d_in[3];        // scalar int (num_batches)
  const int n = in_sizes[2];

  char* ws = (char*)d_ws;
  int* counts = (int*)(ws);                     // MAXB ints
  int* starts = (int*)(ws + 1024);              // MAXB ints
  float* partials = (float*)(ws + 2048);        // NPART floats

  k_counts<<<1, TPB, 0, stream>>>(batch, n, nbp, counts, starts);
  k_focal<<<BLOCKS, TPB, 0, stream>>>(pred, ylab, batch, counts, starts,
                                      partials, n);
  k_final<<<1, TPB, 0, stream>>>(partials, NPART, nbp, (float*)d_out);
}